// ROIAlign_63101659512965
// MI455X (gfx1250) — compile-verified
//
#include <hip/hip_runtime.h>
#include <cstdint>

// Problem constants (from reference setup_inputs / POOL_OUT / SUB)
#define HF   200          // feature height
#define WF   200          // feature width
#define NCH  256          // channels
#define HO   7            // pooled out height
#define WO   7            // pooled out width
#define HS   14           // subsample rows  (HO*SUB)
#define WS   14           // subsample cols  (WO*SUB)
#define NS   (HS * WS)    // 196 subsample points per ROI
#define NPIX (HO * WO)    // 49 outputs per (roi, channel)
#define TPB  256
#define ITERS ((NCH * NPIX) / TPB)   // 12544 / 256 = 49, exact
#define PLANE_BYTES (HF * WF * 4)    // 160000 bytes per channel plane

// Async gather, GVS form: global addr from SGPR base + 32-bit VGPR offset, LDS addr from a VGPR.
// IOFFSET is added to BOTH sides, so we bias the table by (+64 - 4k) and the
// LDS base by +64, and use IOFFSET = 4k - 64 to land tap k in LDS slot k.
#define ASYNC_GATHER(ldsv, voffv, sbase, OFFLIT)                               \
    asm volatile("global_load_async_to_lds_b32 %0, %1, %2 offset:" #OFFLIT     \
                 :: "v"(ldsv), "v"(voffv), "s"(sbase) : "memory")

__global__ __launch_bounds__(TPB)
void ROIAlign_async_kernel(const float* __restrict__ feat,
                           const float* __restrict__ rois,
                           const float* __restrict__ img_size,
                           float* __restrict__ out)
{
    // Per-ROI sample table: 4 biased byte-offsets + 4 bilinear weights / subsample
    __shared__ int4   s_off[NS];                        // 3136 B
    __shared__ float4 s_wgt[NS];                        // 3136 B
    // Double-buffered async gather staging: 16 floats per thread per buffer
    __shared__ __align__(16) float s_buf[2][TPB * 16];  // 32 KB

    const int roi = blockIdx.x;
    const int tid = threadIdx.x;

    // ---- Stage 1: per-ROI sample coordinates / weights (196 threads) ----
    if (tid < NS) {
        const float y1 = rois[roi * 4 + 0], x1 = rois[roi * 4 + 1];
        const float y2 = rois[roi * 4 + 2], x2 = rois[roi * 4 + 3];
        const float Hden = img_size[0] - 1.0f;
        const float Wden = img_size[1] - 1.0f;
        // r = roi * (f-1) / (img-1)  (multiply first, divide second, as reference)
        const float r0 = (y1 * (float)(HF - 1)) / Hden;
        const float r1 = (x1 * (float)(WF - 1)) / Wden;
        const float r2 = (y2 * (float)(HF - 1)) / Hden;
        const float r3 = (x2 * (float)(WF - 1)) / Wden;
        const float hstep = (r2 - r0) / (float)HS;
        const float wstep = (r3 - r1) / (float)WS;

        const int ys = tid / WS;
        const int xs = tid - ys * WS;
        const float y = ((float)ys + 0.5f) * hstep + r0;
        const float x = ((float)xs + 0.5f) * wstep + r1;

        const float yf = floorf(y), xf = floorf(x);
        int iu = (int)yf;
        int id = (int)ceilf(y);
        int il = (int)xf;
        int ir = (int)ceilf(x);
        iu = min(max(iu, 0), HF - 1);  id = min(max(id, 0), HF - 1);
        il = min(max(il, 0), WF - 1);  ir = min(max(ir, 0), WF - 1);
        const float fy = y - yf;
        const float fx = x - xf;

        // This subsample always lands in quadrant (dy,dx) = (ys&1, xs&1) of its
        // output pixel -> its LDS slot group kq = ((ys&1)*2 + (xs&1))*4 is static.
        const int kq   = ((ys & 1) * 2 + (xs & 1)) * 4;
        const int bias = 64 - 4 * kq;   // per-tap t adds an extra -4*t
        s_off[tid] = make_int4((iu * WF + il) * 4 + bias,       // t=0
                               (id * WF + il) * 4 + bias - 4,   // t=1
                               (iu * WF + ir) * 4 + bias - 8,   // t=2
                               (id * WF + ir) * 4 + bias - 12); // t=3
        s_wgt[tid] = make_float4((1.0f - fy) * (1.0f - fx),
                                 fy          * (1.0f - fx),
                                 (1.0f - fy) * fx,
                                 fy          * fx);
    }
    __syncthreads();

    // ---- Stage 2: pipelined async gather + bilinear + 2x2 max ----
    const uint64_t sbase = (uint64_t)(uintptr_t)feat;   // SGPR pair for saddr
    // Low 32 bits of a generic LDS pointer = wave-relative LDS byte address.
    const uint32_t lds0 = (uint32_t)(uintptr_t)&s_buf[0][tid * 16] + 64u;
    const uint32_t lds1 = (uint32_t)(uintptr_t)&s_buf[1][tid * 16] + 64u;

    auto issue_iter = [&](int it, uint32_t ldsb) {
        const int idx = it * TPB + tid;
        const int c   = idx / NPIX;
        const int p   = idx - c * NPIX;
        const int oy  = p / WO;
        const int ox  = p - oy * WO;
        const int s00 = (2 * oy) * WS + 2 * ox;
        const uint32_t cb = (uint32_t)c * (uint32_t)PLANE_BYTES;

        {   // subsample (0,0) -> slots 0..3, IOFFSET = 4k-64 for k=0..3
            const int4 o = s_off[s00];
            ASYNC_GATHER(ldsb, cb + (uint32_t)o.x, sbase, -64);
            ASYNC_GATHER(ldsb, cb + (uint32_t)o.y, sbase, -60);
            ASYNC_GATHER(ldsb, cb + (uint32_t)o.z, sbase, -56);
            ASYNC_GATHER(ldsb, cb + (uint32_t)o.w, sbase, -52);
        }
        {   // subsample (0,1) -> slots 4..7
            const int4 o = s_off[s00 + 1];
            ASYNC_GATHER(ldsb, cb + (uint32_t)o.x, sbase, -48);
            ASYNC_GATHER(ldsb, cb + (uint32_t)o.y, sbase, -44);
            ASYNC_GATHER(ldsb, cb + (uint32_t)o.z, sbase, -40);
            ASYNC_GATHER(ldsb, cb + (uint32_t)o.w, sbase, -36);
        }
        {   // subsample (1,0) -> slots 8..11
            const int4 o = s_off[s00 + WS];
            ASYNC_GATHER(ldsb, cb + (uint32_t)o.x, sbase, -32);
            ASYNC_GATHER(ldsb, cb + (uint32_t)o.y, sbase, -28);
            ASYNC_GATHER(ldsb, cb + (uint32_t)o.z, sbase, -24);
            ASYNC_GATHER(ldsb, cb + (uint32_t)o.w, sbase, -20);
        }
        {   // subsample (1,1) -> slots 12..15
            const int4 o = s_off[s00 + WS + 1];
            ASYNC_GATHER(ldsb, cb + (uint32_t)o.x, sbase, -16);
            ASYNC_GATHER(ldsb, cb + (uint32_t)o.y, sbase, -12);
            ASYNC_GATHER(ldsb, cb + (uint32_t)o.z, sbase, -8);
            ASYNC_GATHER(ldsb, cb + (uint32_t)o.w, sbase, -4);
        }
    };

    issue_iter(0, lds0);   // prime the pipeline (16 async gathers in flight)

    for (int it = 0; it < ITERS; ++it) {
        const int b = it & 1;
        if (it + 1 < ITERS) {
            issue_iter(it + 1, b ? lds0 : lds1);      // now <= 32 in flight
            // async loads complete in order -> waiting to <=16 guarantees the
            // 16 gathers for iteration `it` have landed in s_buf[b]
            asm volatile("s_wait_asynccnt 16" ::: "memory");
        } else {
            asm volatile("s_wait_asynccnt 0" ::: "memory");
        }

        const int idx = it * TPB + tid;
        const int c   = idx / NPIX;
        const int p   = idx - c * NPIX;
        const int oy  = p / WO;
        const int ox  = p - oy * WO;
        const int s00 = (2 * oy) * WS + 2 * ox;
        const float4* buf4 = (const float4*)&s_buf[b][tid * 16];

        float m = -3.402823466e38f;
#pragma unroll
        for (int dy = 0; dy < 2; ++dy) {
#pragma unroll
            for (int dx = 0; dx < 2; ++dx) {
                const float4 w = s_wgt[s00 + dy * WS + dx];
                const float4 v = buf4[dy * 2 + dx];
                const float acc = v.x * w.x + v.y * w.y + v.z * w.z + v.w * w.w;
                m = fmaxf(m, acc);
            }
        }
        out[(size_t)(roi * NCH + c) * NPIX + p] = m;
    }
}

extern "C" void kernel_launch(void* const* d_in, const int* in_sizes, int n_in,
                              void* d_out, int out_size, void* d_ws, size_t ws_size,
                              hipStream_t stream)
{
    const float* feat = (const float*)d_in[0];   // (1, 256, 200, 200) f32
    const float* rois = (const float*)d_in[1];   // (N, 4) f32
    const float* img  = (const float*)d_in[2];   // (2,)  f32
    float* out = (float*)d_out;                  // (N, 256, 7, 7) f32

    const int n_roi = in_sizes[1] / 4;
    ROIAlign_async_kernel<<<n_roi, TPB, 0, stream>>>(feat, rois, img, out);
}